// SoftmaxBernoulli2_82884278879164
// MI455X (gfx1250) — compile-verified
//
#include <hip/hip_runtime.h>
#include <stdint.h>

// Quantized softmax (horizon SoftmaxBernoulli2) over rows of 2048 int32.
// x: (2*16*2048) rows x 2048, int32 in [-128,128). out: f32 same shape.
// Pure-bandwidth kernel: 512MB in + 512MB out => ~46us at 23.3 TB/s.
// Row staged in LDS via CDNA5 async global->LDS DMA (ASYNCcnt), overlapped
// with on-the-fly construction of the 256-entry exp LUT in LDS.

#define ROW   2048
#define TPB   256
#define EPT   (ROW / TPB)   // 8 elements per thread
#define NWAVE (TPB / 32)    // 8 wave32 waves per block

typedef __attribute__((ext_vector_type(4))) int   i4;
typedef __attribute__((ext_vector_type(4))) float f4;

__global__ __launch_bounds__(TPB) void softmax_q8_row2048(
    const int*   __restrict__ x,
    const float* __restrict__ input_scale,
    const float* __restrict__ exp_scale,
    float*       __restrict__ out)
{
    __shared__ alignas(16) int lds_row[ROW];   // 8 KB: one full row
    __shared__ int lds_lut[256];               // 1 KB: exp LUT
    __shared__ int lds_red[NWAVE];             // cross-wave reduction scratch

    const int    tid  = (int)threadIdx.x;
    const size_t base = (size_t)blockIdx.x * (size_t)ROW;

    // ---- 1) Kick off async global->LDS copy of this row: 32 B per thread ----
    // CDNA5: GLOBAL_LOAD_ASYNC_TO_LDS_B128, tracked by ASYNCcnt. The 24-bit
    // instruction offset is added to BOTH the LDS and the global address, so
    // one (lds,global) register pair covers both 16B halves.
    {
        uint64_t ga = (uint64_t)(uintptr_t)(x + base) + (uint64_t)(tid * 32);
        uint32_t la = (uint32_t)(uintptr_t)lds_row + (uint32_t)(tid * 32);
        asm volatile(
            "global_load_async_to_lds_b128 %0, %1, off\n\t"
            "global_load_async_to_lds_b128 %0, %1, off offset:16"
            :
            : "v"(la), "v"(ga)
            : "memory");
    }

    // ---- 2) Build exp LUT while the DMA is in flight ----
    // table[q+255] = clip(rint(exp(q*input_scale)/exp_scale), 0, 255), q in [-255,0].
    // rintf == round-half-to-even == jnp.round.
    {
        const float is = input_scale[0];
        const float es = exp_scale[0];
        float q = (float)(tid - 255) * is;
        float t = rintf(expf(q) / es);
        t = fminf(fmaxf(t, 0.0f), 255.0f);
        lds_lut[tid] = (int)t;
    }

    // ---- 3) Wait for our own async ops, then barrier (LUT visibility) ----
#if __has_builtin(__builtin_amdgcn_s_wait_asynccnt)
    __builtin_amdgcn_s_wait_asynccnt(0);
#else
    asm volatile("s_wait_asynccnt 0" ::: "memory");
#endif
    __syncthreads();

    // ---- 4) Read back own 8 elements (two ds_load_b128), row max ----
    int v[EPT];
    {
        const i4 a = *(const i4*)&lds_row[tid * EPT];
        const i4 b = *(const i4*)&lds_row[tid * EPT + 4];
        v[0] = a.x; v[1] = a.y; v[2] = a.z; v[3] = a.w;
        v[4] = b.x; v[5] = b.y; v[6] = b.z; v[7] = b.w;
    }

    int m = v[0];
#pragma unroll
    for (int k = 1; k < EPT; ++k) m = max(m, v[k]);
#pragma unroll
    for (int off = 16; off > 0; off >>= 1) m = max(m, __shfl_xor(m, off, 32));
    if ((tid & 31) == 0) lds_red[tid >> 5] = m;
    __syncthreads();
    m = lds_red[0];
#pragma unroll
    for (int i = 1; i < NWAVE; ++i) m = max(m, lds_red[i]);
    __syncthreads();   // WAR guard before lds_red is reused for the sum

    // ---- 5) LUT gather + integer row sum ----
    int e[EPT];
    int s = 0;
#pragma unroll
    for (int k = 0; k < EPT; ++k) {
        int idx = (v[k] - m) + 255;        // in [0,255] since |x-max| <= 255
        idx = min(max(idx, 0), 255);
        e[k] = lds_lut[idx];
        s += e[k];
    }
#pragma unroll
    for (int off = 16; off > 0; off >>= 1) s += __shfl_xor(s, off, 32);
    if ((tid & 31) == 0) lds_red[tid >> 5] = s;
    __syncthreads();
    s = lds_red[0];
#pragma unroll
    for (int i = 1; i < NWAVE; ++i) s += lds_red[i];

    // ---- 6) Normalize and stream out (non-temporal b128 stores) ----
    const float inv = 1.0f / (float)s;     // one IEEE divide per thread
    f4 o0, o1;
    o0.x = (float)e[0] * inv; o0.y = (float)e[1] * inv;
    o0.z = (float)e[2] * inv; o0.w = (float)e[3] * inv;
    o1.x = (float)e[4] * inv; o1.y = (float)e[5] * inv;
    o1.z = (float)e[6] * inv; o1.w = (float)e[7] * inv;

    float* op = out + base + (size_t)(tid * EPT);
    __builtin_nontemporal_store(o0, (f4*)op);
    __builtin_nontemporal_store(o1, (f4*)(op + 4));
}

extern "C" void kernel_launch(void* const* d_in, const int* in_sizes, int n_in,
                              void* d_out, int out_size, void* d_ws, size_t ws_size,
                              hipStream_t stream) {
    (void)n_in; (void)out_size; (void)d_ws; (void)ws_size;
    const int*   x  = (const int*)d_in[0];
    const float* is = (const float*)d_in[1];
    const float* es = (const float*)d_in[2];
    float*       o  = (float*)d_out;

    const unsigned rows = (unsigned)(in_sizes[0] / ROW);   // 2*16*2048 = 65536
    softmax_q8_row2048<<<dim3(rows), dim3(TPB), 0, stream>>>(x, is, es, o);
}